// CRF_73315091742818
// MI455X (gfx1250) — compile-verified
//
#include <hip/hip_runtime.h>
#include <hip/hip_bf16.h>

typedef __attribute__((ext_vector_type(2))) float v2f;
typedef __attribute__((ext_vector_type(8))) float v8f;

#define S_LEN 512
#define B_DIM 128
#define H_DIM 512
#define T_TAG 48

// ---------------------------------------------------------------------------
// Kernel 1: emissions[s,b,t] = word_features[s,b,:] . W[t,:] + bias[t]
// M = S*B = 65536, K = H = 512, N = T = 48, all fp32 via V_WMMA_F32_16X16X4_F32.
// One wave handles one 16-row M tile across all 3 N tiles (A fragment reused 3x).
// Block = 256 threads = 8 waves -> grid = 65536 / (16*8) = 512 blocks.
// ---------------------------------------------------------------------------
__global__ __launch_bounds__(256) void emis_gemm_wmma(
    const float* __restrict__ A,      // [M, H] = word_features flattened
    const float* __restrict__ W,      // [T, H]
    const float* __restrict__ bias,   // [T]
    float* __restrict__ out)          // [M, T]
{
    const int lane = threadIdx.x & 31;
    const int wave = threadIdx.x >> 5;
    const int m0   = (blockIdx.x * 8 + wave) << 4;   // 16-row tile base
    const int l16  = lane & 15;
    const int kh   = lane >> 4;                      // lane-half selects K pair

    // A fragment: lane holds row m0+l16; VGPR0/1 = K = (2*kh)+0 / +1  (float2)
    const float* arow = A + (size_t)(m0 + l16) * H_DIM + 2 * kh;
    // B fragment (B = W^T, K x N): lane holds column n = n0+l16, same K split
    const float* wr0  = W + (size_t)( 0 + l16) * H_DIM + 2 * kh;
    const float* wr1  = W + (size_t)(16 + l16) * H_DIM + 2 * kh;
    const float* wr2  = W + (size_t)(32 + l16) * H_DIM + 2 * kh;

    v8f acc0 = {}, acc1 = {}, acc2 = {};
    #pragma unroll 4
    for (int k = 0; k < H_DIM; k += 4) {
        v2f a  = *(const v2f*)(arow + k);
        v2f b0 = *(const v2f*)(wr0 + k);
        v2f b1 = *(const v2f*)(wr1 + k);
        v2f b2 = *(const v2f*)(wr2 + k);
        acc0 = __builtin_amdgcn_wmma_f32_16x16x4_f32(false, a, false, b0, (short)0, acc0, false, false);
        acc1 = __builtin_amdgcn_wmma_f32_16x16x4_f32(false, a, false, b1, (short)0, acc1, false, false);
        acc2 = __builtin_amdgcn_wmma_f32_16x16x4_f32(false, a, false, b2, (short)0, acc2, false, false);
    }

    const float bb0 = bias[ 0 + l16];
    const float bb1 = bias[16 + l16];
    const float bb2 = bias[32 + l16];
    #pragma unroll
    for (int r = 0; r < 8; ++r) {
        const int m = m0 + r + 8 * kh;               // C/D layout: M = r + 8*half
        float* o = out + (size_t)m * T_TAG;
        o[ 0 + l16] = acc0[r] + bb0;
        o[16 + l16] = acc1[r] + bb1;
        o[32 + l16] = acc2[r] + bb2;
    }
}

// ---------------------------------------------------------------------------
// Kernel 2: forward-algorithm scan + gold score, one block per batch element.
// new_alpha[j] = m + log( sum_i exp(alpha[i]-m) * E[i][j] ) + emit[s,j],
// with E = exp(transitions) precomputed in LDS (constant across steps).
// ---------------------------------------------------------------------------
__global__ __launch_bounds__(64) void crf_scan(
    const float* __restrict__ emis,   // [S, B, T]
    const float* __restrict__ trans,  // [T, T]
    const int*   __restrict__ tags,   // [S, B]
    float* __restrict__ partial)      // [B] : logZ - gold
{
    __shared__ float E[T_TAG * T_TAG];
    __shared__ float alpha[T_TAG];
    __shared__ float p[T_TAG];
    __shared__ float red[64];

    const int b   = blockIdx.x;
    const int tid = threadIdx.x;

    // Precompute E = exp(transitions) once (full-precision expf; cost is trivial)
    for (int i = tid; i < T_TAG * T_TAG; i += 64) E[i] = expf(trans[i]);
    if (tid < T_TAG) alpha[tid] = emis[(size_t)b * T_TAG + tid];

    // Gold score partials (independent of the scan)
    float g = 0.0f;
    for (int s = tid; s < S_LEN; s += 64) {
        const int ts = tags[s * B_DIM + b];
        g += emis[((size_t)s * B_DIM + b) * T_TAG + ts];
        if (s > 0) {
            const int tp = tags[(s - 1) * B_DIM + b];
            g += trans[tp * T_TAG + ts];
        }
    }
    red[tid] = g;
    __syncthreads();

    // Serial scan over time
    for (int s = 1; s < S_LEN; ++s) {
        const float* em = emis + ((size_t)s * B_DIM + b) * T_TAG;
        if (tid == 0) {
            const int sp = (s + 8 < S_LEN) ? s + 8 : s;
            __builtin_prefetch(emis + ((size_t)sp * B_DIM + b) * T_TAG, 0, 0);
        }
        // row max (redundant per-thread scan over LDS broadcast reads)
        float m = alpha[0];
        #pragma unroll 8
        for (int i = 1; i < T_TAG; ++i) m = fmaxf(m, alpha[i]);
        if (tid < T_TAG) p[tid] = __expf(alpha[tid] - m);
        __syncthreads();
        if (tid < T_TAG) {
            float sum = 0.0f;
            #pragma unroll 8
            for (int i = 0; i < T_TAG; ++i) sum = fmaf(p[i], E[i * T_TAG + tid], sum);
            alpha[tid] = m + __logf(sum) + em[tid];
        }
        __syncthreads();
    }

    // reduce gold partials
    for (int off = 32; off > 0; off >>= 1) {
        if (tid < off) red[tid] += red[tid + off];
        __syncthreads();
    }

    if (tid == 0) {
        float m = alpha[0];
        for (int i = 1; i < T_TAG; ++i) m = fmaxf(m, alpha[i]);
        float sum = 0.0f;
        for (int i = 0; i < T_TAG; ++i) sum += __expf(alpha[i] - m);
        const float logZ = m + __logf(sum);
        partial[b] = logZ - red[0];
    }
}

// ---------------------------------------------------------------------------
// Kernel 3: nll = mean(partial)  (deterministic tree reduce, no fp atomics)
// ---------------------------------------------------------------------------
__global__ __launch_bounds__(128) void final_reduce(
    const float* __restrict__ partial, float* __restrict__ out)
{
    __shared__ float red[128];
    const int tid = threadIdx.x;
    red[tid] = partial[tid];
    __syncthreads();
    for (int off = 64; off > 0; off >>= 1) {
        if (tid < off) red[tid] += red[tid + off];
        __syncthreads();
    }
    if (tid == 0) out[0] = red[0] * (1.0f / (float)B_DIM);
}

extern "C" void kernel_launch(void* const* d_in, const int* in_sizes, int n_in,
                              void* d_out, int out_size, void* d_ws, size_t ws_size,
                              hipStream_t stream) {
    (void)in_sizes; (void)n_in; (void)out_size; (void)ws_size;

    const float* word_features = (const float*)d_in[0]; // [S,B,H]
    const float* W             = (const float*)d_in[1]; // [T,H]
    const float* bias          = (const float*)d_in[2]; // [T]
    const float* trans         = (const float*)d_in[3]; // [T,T]
    const int*   tags          = (const int*)  d_in[4]; // [S,B]

    float* emis    = (float*)d_ws;                                   // S*B*T floats
    float* partial = emis + (size_t)S_LEN * B_DIM * T_TAG;           // B floats

    emis_gemm_wmma<<<(S_LEN * B_DIM) / (16 * 8), 256, 0, stream>>>(
        word_features, W, bias, emis);
    crf_scan<<<B_DIM, 64, 0, stream>>>(emis, trans, tags, partial);
    final_reduce<<<1, 128, 0, stream>>>(partial, (float*)d_out);
}